// PyramidRenderer_49220325212762
// MI455X (gfx1250) — compile-verified
//
#include <hip/hip_runtime.h>
#include <stdint.h>

// NeRF volume rendering, MI455X (gfx1250).
// Memory-bound: ~405 MB total traffic -> ~17us floor at 23.3 TB/s.
// Strategy: one wave32 per ray (fully coalesced row access), shuffle-based
// exclusive cumprod scan (6 chunks of 32 samples with a scalar carry),
// and gfx1250 async global->LDS staging (ASYNCcnt) for the rgb stream:
// one b128 async DMA per chunk (lanes 0-23 x 16B = 384B), double-buffered
// so the DMA overlaps the exp/scan ALU work. One-shot streams (sigma in,
// weights/trans/alpha out) use non-temporal hints to spare the 192MB L2
// for the z/rgb reuse windows.

#define N_RAYS  65536
#define N_SAMP  192
#define WPB     8                 // waves per block (256 threads)
#define CHUNKS  (N_SAMP / 32)     // 6

__global__ __launch_bounds__(WPB * 32)
void nerf_render_kernel(const float* __restrict__ rgb,
                        const float* __restrict__ sigma,
                        const float* __restrict__ zvals,
                        const float* __restrict__ rays_d,
                        float* __restrict__ out)
{
    // double-buffered rgb staging: 96 dwords (32 samples * 3ch) per chunk.
    // 16B-aligned so the b128 async DMA lands on native alignment.
    __shared__ alignas(16) float smem[WPB][2][96];

    const int lane = threadIdx.x & 31;
    const int widx = threadIdx.x >> 5;
    const int ray  = blockIdx.x * WPB + widx;   // wave-uniform
    if (ray >= N_RAYS) return;

    const float* zrow = zvals + (size_t)ray * N_SAMP;
    const float* srow = sigma + (size_t)ray * N_SAMP;

    // Wave-uniform rgb row base, forced scalar so the async SADDR is an SGPR pair.
    const int rayU = __builtin_amdgcn_readfirstlane(ray);
    const float* growS = rgb + (size_t)rayU * N_SAMP * 3;

    // Output layout: tuple concatenated flat in return order.
    const size_t N  = N_RAYS;
    const size_t NS = (size_t)N_RAYS * N_SAMP;
    float* out_rgb   = out;                 // [N,3]
    float* out_depth = out + 3 * N;         // [N]
    float* out_acc   = out + 4 * N;         // [N]
    float* out_w     = out + 5 * N;         // [N,S]
    float* out_disp  = out + 5 * N + NS;    // [N]
    float* out_t     = out + 6 * N + NS;    // [N,S]
    float* out_a     = out + 6 * N + 2*NS;  // [N,S]

    // |rays_d| (broadcast loads, uniform per wave)
    const float dx = rays_d[3 * ray + 0];
    const float dy = rays_d[3 * ray + 1];
    const float dz = rays_d[3 * ray + 2];
    const float nrm = sqrtf(dx * dx + dy * dy + dz * dz);

    // LDS byte offset of this wave's staging area (generic ptr low 32 bits).
    const unsigned smemBase = (unsigned)(uintptr_t)&smem[widx][0][0];

    // Stage chunk c's 384B of rgb into LDS buffer `buf` with a single b128
    // async DMA op (GVS addressing: SGPR base + VGPR offset), lanes 0-23.
    auto stage = [&](int c, int buf) {
        if (lane < 24) {
            unsigned lo = smemBase + (unsigned)(buf * 384 + lane * 16);
            unsigned go = (unsigned)(c * 384 + lane * 16);
            asm volatile("global_load_async_to_lds_b128 %0, %1, %2"
                         :: "v"(lo), "v"(go), "s"(growS)
                         : "memory");
        }
    };

    stage(0, 0);

    float carry = 1.0f;   // running transmittance across chunks
    float acc_r = 0.f, acc_g = 0.f, acc_b = 0.f, acc_d = 0.f, acc_w = 0.f;

#pragma unroll
    for (int c = 0; c < CHUNKS; ++c) {
        const int i = c * 32 + lane;

        const float z  = zrow[i];
        const float zn = (i < N_SAMP - 1) ? zrow[i + 1] : 0.0f;
        const float dist = ((i == N_SAMP - 1) ? 1e10f : (zn - z)) * nrm;
        const float s  = fmaxf(__builtin_nontemporal_load(&srow[i]), 0.0f);
        const float alpha = 1.0f - expf(-s * dist);
        const float om = 1.0f - alpha;

        // inclusive product scan of (1-alpha) across the wave (5 shuffle steps)
        float p = om;
#pragma unroll
        for (int off = 1; off < 32; off <<= 1) {
            const float v = __shfl_up(p, off);
            if (lane >= off) p *= v;
        }
        float excl = __shfl_up(p, 1);
        if (lane == 0) excl = 1.0f;

        const float trans = carry * excl;   // T_i = prod_{j<i}(1-alpha_j)
        const float w     = alpha * trans;
        carry *= __shfl(p, 31);             // fold full-chunk product into carry

        const size_t o = (size_t)ray * N_SAMP + i;
        __builtin_nontemporal_store(w,     &out_w[o]);
        __builtin_nontemporal_store(trans, &out_t[o]);
        __builtin_nontemporal_store(alpha, &out_a[o]);

        // Kick next chunk's DMA, then retire the buffer we're about to read.
        if (c + 1 < CHUNKS) {
            stage(c + 1, (c + 1) & 1);
            asm volatile("s_wait_asynccnt 0x1" ::: "memory");  // oldest done
        } else {
            asm volatile("s_wait_asynccnt 0x0" ::: "memory");
        }

        // rgb for this sample from LDS (stride-3: bank-conflict-free)
        const float* cb = &smem[widx][c & 1][0];
        const float r = cb[3 * lane + 0];
        const float g = cb[3 * lane + 1];
        const float b = cb[3 * lane + 2];

        acc_r += w * r;  acc_g += w * g;  acc_b += w * b;
        acc_d += w * z;  acc_w += w;
    }

    // wave-level tree reduction of the 5 accumulators
#pragma unroll
    for (int off = 16; off > 0; off >>= 1) {
        acc_r += __shfl_down(acc_r, off);
        acc_g += __shfl_down(acc_g, off);
        acc_b += __shfl_down(acc_b, off);
        acc_d += __shfl_down(acc_d, off);
        acc_w += __shfl_down(acc_w, off);
    }

    if (lane == 0) {
        out_rgb[3 * ray + 0] = acc_r;
        out_rgb[3 * ray + 1] = acc_g;
        out_rgb[3 * ray + 2] = acc_b;
        out_depth[ray] = acc_d;
        out_acc[ray]   = acc_w;
        out_disp[ray]  = 1.0f / fmaxf(1e-10f, acc_d / acc_w);
    }
}

extern "C" void kernel_launch(void* const* d_in, const int* in_sizes, int n_in,
                              void* d_out, int out_size, void* d_ws, size_t ws_size,
                              hipStream_t stream) {
    (void)in_sizes; (void)n_in; (void)out_size; (void)d_ws; (void)ws_size;
    const float* rgb    = (const float*)d_in[0];
    const float* sigma  = (const float*)d_in[1];
    const float* zvals  = (const float*)d_in[2];
    const float* rays_d = (const float*)d_in[3];
    float* out = (float*)d_out;

    dim3 grid(N_RAYS / WPB);   // 8192 blocks
    dim3 block(WPB * 32);      // 256 threads = 8 wave32
    nerf_render_kernel<<<grid, block, 0, stream>>>(rgb, sigma, zvals, rays_d, out);
}